// PointNetSAModule_22162031247559
// MI455X (gfx1250) — compile-verified
//
#include <hip/hip_runtime.h>

// Problem constants (match reference: B=4, N=16384, K=16, C=64, H=128, IN_DIM=131)
static constexpr int Bb   = 4;
static constexpr int Nn   = 16384;
static constexpr int KNN  = 16;
static constexpr int Cc   = 64;
static constexpr int Hh   = 128;
static constexpr int IND  = 131;          // 3 + 2C
static constexpr int KP   = 132;          // padded K / LDS row stride (floats) -> conflict-free frags
static constexpr int PTS_PER_WG = 8;      // 8 waves, one point (16 rows) per wave
static constexpr int ROWS = PTS_PER_WG * KNN;   // 128 rows of x per workgroup
static constexpr int THREADS = 256;

typedef __attribute__((ext_vector_type(2))) float v2f;
typedef __attribute__((ext_vector_type(8))) float v8f;

__device__ __forceinline__ v8f splat8(float v) {
  v8f r = {v, v, v, v, v, v, v, v};
  return r;
}

// Stage W1 transposed + k-permuted into LDS: wt[n*KP + kp] = W1[k][n]
// perm: x layout is [central feats 0..63 | nb feats 64..127 | xyz 128..130 | pad 131]
__device__ __forceinline__ void stage_w1(float* wt, const float* __restrict__ W1, int tid) {
  for (int e = tid; e < IND * Hh; e += THREADS) {
    int k = e >> 7;             // row of W1 (0..130)
    int n = e & (Hh - 1);       // col of W1 (0..127)
    int kp = (k < 3) ? (128 + k) : (k - 3);
    wt[n * KP + kp] = W1[e];
  }
  for (int n = tid; n < Hh; n += THREADS) wt[n * KP + 131] = 0.0f;  // pad column
}

// Cooperative gather/build of the x tile: 2 threads per row, float4 feature copies.
__device__ __forceinline__ void build_x(float* xs, const float* __restrict__ xyz,
                                        const float* __restrict__ feat,
                                        const long long* __restrict__ nb,
                                        int point0, int tid) {
  const int row  = tid >> 1;        // 0..127
  const int half = tid & 1;
  const int lp   = row >> 4;        // local point 0..7
  const int j    = row & 15;        // neighbor slot
  const int P    = point0 + lp;     // global point id in [0, B*N)
  const int bIdx = P >> 14;         // N == 16384
  const long long idx = nb[(long long)P * KNN + j];   // int64 neighbor index
  const int nbP = bIdx * Nn + (int)idx;
  float* xr = xs + row * KP;
  if (half == 0) {
    const float4* fc = (const float4*)(feat + (long long)P * Cc);
    float4* dst = (float4*)xr;                     // 16B aligned (KP*4 = 528)
#pragma unroll
    for (int q = 0; q < Cc / 4; ++q) dst[q] = fc[q];
    xr[128] = xyz[nbP * 3 + 0] - xyz[P * 3 + 0];
    xr[129] = xyz[nbP * 3 + 1] - xyz[P * 3 + 1];
    xr[130] = xyz[nbP * 3 + 2] - xyz[P * 3 + 2];
    xr[131] = 0.0f;
  } else {
    const float4* fn = (const float4*)(feat + (long long)nbP * Cc);
    float4* dst = (float4*)(xr + Cc);
#pragma unroll
    for (int q = 0; q < Cc / 4; ++q) dst[q] = fn[q];
  }
}

// One wave: 16-row A tile (aBase) x staged weights (wt) -> 8 accumulators of 16x16.
// A frag: lanes 0-15 read [r=lane][k0,k0+1]; lanes 16-31 read [r=lane-16][k0+2,k0+3].
// B frag: same pattern on wt rows (wt holds W transposed: row n, col k).
__device__ __forceinline__ void gemm_tile(v8f acc[8], const float* aBase, const float* wt,
                                          int ksteps, int lq, int koff) {
  for (int ks = 0; ks < ksteps; ++ks) {
    const int k0 = ks * 4;
    v2f a = *(const v2f*)(aBase + k0);
#pragma unroll
    for (int nt = 0; nt < 8; ++nt) {
      v2f bf = *(const v2f*)(wt + (nt * 16 + lq) * KP + k0 + koff);
      acc[nt] = __builtin_amdgcn_wmma_f32_16x16x4_f32(
          /*neg_a=*/false, a, /*neg_b=*/false, bf,
          /*c_mod=*/(short)0, acc[nt], /*reuse_a=*/false, /*reuse_b=*/false);
    }
  }
}

// ---------------- Pass 1: per-channel sum / sum-of-squares of h = x@W1 + b1 ----------------
extern "C" __global__ void __launch_bounds__(THREADS)
pn_stats(const float* __restrict__ xyz, const float* __restrict__ feat,
         const long long* __restrict__ nb, const float* __restrict__ W1,
         const float* __restrict__ b1, float* __restrict__ ws) {
  extern __shared__ float smem[];
  float* xs  = smem;                     // ROWS*KP
  float* wt  = smem + ROWS * KP;         // Hh*KP
  float* red = wt + Hh * KP;             // 2*Hh reduction scratch

  const int tid  = threadIdx.x;
  const int lane = tid & 31;
  const int wv   = tid >> 5;
  const int point0 = blockIdx.x * PTS_PER_WG;

  stage_w1(wt, W1, tid);
  build_x(xs, xyz, feat, nb, point0, tid);
  if (tid < 2 * Hh) red[tid] = 0.0f;
  __syncthreads();

  const int lq   = lane & 15;
  const int koff = (lane < 16) ? 0 : 2;
  const float* aBase = xs + (wv * 16 + lq) * KP + koff;

  v8f acc[8];
#pragma unroll
  for (int nt = 0; nt < 8; ++nt) acc[nt] = splat8(b1[nt * 16 + lq]);
  gemm_tile(acc, aBase, wt, 33, lq, koff);   // 131 (+pad) = 33 k-steps

#pragma unroll
  for (int nt = 0; nt < 8; ++nt) {
    float s = 0.0f, s2 = 0.0f;
#pragma unroll
    for (int j = 0; j < 8; ++j) { float v = acc[nt][j]; s += v; s2 += v * v; }
    s  += __shfl_xor(s, 16, 32);
    s2 += __shfl_xor(s2, 16, 32);
    if (lane < 16) {
      atomicAdd(&red[nt * 16 + lq], s);          // ds_add_f32
      atomicAdd(&red[Hh + nt * 16 + lq], s2);
    }
  }
  __syncthreads();
  if (tid < 2 * Hh) atomicAdd(&ws[tid], red[tid]);   // one global atomic per channel per WG
}

// ---------------- Fold BN stats + gamma/beta into per-channel scale/shift ----------------
extern "C" __global__ void pn_finalize(const float* __restrict__ gamma,
                                       const float* __restrict__ beta,
                                       float* __restrict__ ws) {
  const int c = threadIdx.x;             // 0..127
  const float invM = 1.0f / (float)((long long)Bb * Nn * KNN);
  const float mean = ws[c] * invM;
  const float var  = ws[Hh + c] * invM - mean * mean;
  const float sc   = gamma[c] * rsqrtf(var + 1e-5f);
  ws[2 * Hh + c] = sc;
  ws[3 * Hh + c] = beta[c] - mean * sc;
}

// ---------------- Pass 2: h = BN(x@W1+b1); relu; (h@W2+b2); max over K ----------------
extern "C" __global__ void __launch_bounds__(THREADS)
pn_main(const float* __restrict__ xyz, const float* __restrict__ feat,
        const long long* __restrict__ nb, const float* __restrict__ W1,
        const float* __restrict__ b1, const float* __restrict__ W2,
        const float* __restrict__ b2, const float* __restrict__ ws,
        float* __restrict__ out) {
  extern __shared__ float smem[];
  float* xs = smem;                 // ROWS*KP: x tile, later per-wave h tiles
  float* wt = smem + ROWS * KP;     // Hh*KP:  W1^T, later W2^T

  const int tid  = threadIdx.x;
  const int lane = tid & 31;
  const int wv   = tid >> 5;
  const int point0 = blockIdx.x * PTS_PER_WG;
  const int lq   = lane & 15;
  const int koff = (lane < 16) ? 0 : 2;

  float scf[8], shf[8], b2f[8];
#pragma unroll
  for (int nt = 0; nt < 8; ++nt) {
    const int c = nt * 16 + lq;
    scf[nt] = ws[2 * Hh + c];
    shf[nt] = ws[3 * Hh + c];
    b2f[nt] = b2[c];
  }

  stage_w1(wt, W1, tid);
  build_x(xs, xyz, feat, nb, point0, tid);
  __syncthreads();

  const float* aBase = xs + (wv * 16 + lq) * KP + koff;

  v8f acc[8];
#pragma unroll
  for (int nt = 0; nt < 8; ++nt) acc[nt] = splat8(b1[nt * 16 + lq]);
  gemm_tile(acc, aBase, wt, 33, lq, koff);

  // BN + ReLU, write h into this wave's own xs slice (C/D layout -> row-major LDS)
  float* hBase = xs + (wv * 16) * KP;
  const int mAdd = (lane < 16) ? 0 : 8;
#pragma unroll
  for (int nt = 0; nt < 8; ++nt) {
#pragma unroll
    for (int j = 0; j < 8; ++j) {
      float v = fmaxf(acc[nt][j] * scf[nt] + shf[nt], 0.0f);
      hBase[(j + mAdd) * KP + nt * 16 + lq] = v;
    }
  }
  __syncthreads();                       // all waves done with W1 region

  // Restage wt with W2^T (k = 0..127; pad cols never read since 32 k-steps)
  for (int e = tid; e < Hh * Hh; e += THREADS) {
    int k = e >> 7, n = e & (Hh - 1);
    wt[n * KP + k] = W2[e];
  }
  __syncthreads();

#pragma unroll
  for (int nt = 0; nt < 8; ++nt) acc[nt] = splat8(b2f[nt]);
  gemm_tile(acc, aBase, wt, 32, lq, koff);   // h is 128 wide -> 32 k-steps

  // Max-pool over the wave's 16 rows (K neighbors) and store (B,N,H)
  const int P = point0 + wv;
#pragma unroll
  for (int nt = 0; nt < 8; ++nt) {
    float m = acc[nt][0];
#pragma unroll
    for (int j = 1; j < 8; ++j) m = fmaxf(m, acc[nt][j]);
    m = fmaxf(m, __shfl_xor(m, 16, 32));     // combine rows 0-7 with rows 8-15
    if (lane < 16) out[(long long)P * Hh + nt * 16 + lq] = m;
  }
}

extern "C" void kernel_launch(void* const* d_in, const int* in_sizes, int n_in,
                              void* d_out, int out_size, void* d_ws, size_t ws_size,
                              hipStream_t stream) {
  const float*     xyz   = (const float*)d_in[0];
  const float*     feat  = (const float*)d_in[1];
  const long long* nbidx = (const long long*)d_in[2];   // int64 per reference
  const float*     W1    = (const float*)d_in[3];
  const float*     b1    = (const float*)d_in[4];
  const float*     gamma = (const float*)d_in[5];
  const float*     beta  = (const float*)d_in[6];
  const float*     W2    = (const float*)d_in[7];
  const float*     b2    = (const float*)d_in[8];
  float* out = (float*)d_out;
  float* ws  = (float*)d_ws;

  // zero the sum/sumsq accumulators (ws is poisoned by the harness)
  hipMemsetAsync(ws, 0, 2 * Hh * sizeof(float), stream);

  const int nWG = (Bb * Nn) / PTS_PER_WG;   // 8192 workgroups
  const size_t smemStats = (size_t)(ROWS * KP + Hh * KP + 2 * Hh) * sizeof(float); // ~133 KB
  const size_t smemMain  = (size_t)(ROWS * KP + Hh * KP) * sizeof(float);          // ~132 KB

  pn_stats<<<nWG, THREADS, smemStats, stream>>>(xyz, feat, nbidx, W1, b1, ws);
  pn_finalize<<<1, Hh, 0, stream>>>(gamma, beta, ws);
  pn_main<<<nWG, THREADS, smemMain, stream>>>(xyz, feat, nbidx, W1, b1, W2, b2, ws, out);
}